// LinearTransformer_86268713107944
// MI455X (gfx1250) — compile-verified
//
#include <hip/hip_runtime.h>
#include <hip/hip_bf16.h>

#define N_IN  4096
#define N_HID 4096
#define N_OUT 4096
#define NJB   64          // number of 64-wide j blocks

typedef float v2f __attribute__((ext_vector_type(2)));
typedef float v8f __attribute__((ext_vector_type(8)));

static __device__ __forceinline__ v8f wmma_f32_16x16x4(v2f a, v2f b, v8f c) {
#if defined(__has_builtin) && __has_builtin(__builtin_amdgcn_wmma_f32_16x16x4_f32)
  // 8 args: (neg_a, A, neg_b, B, c_mod, C, reuse_a, reuse_b)
  return __builtin_amdgcn_wmma_f32_16x16x4_f32(false, a, false, b, (short)0, c,
                                               false, false);
#else
  v8f d;
  asm volatile("v_wmma_f32_16x16x4_f32 %0, %1, %2, %3"
               : "=v"(d) : "v"(a), "v"(b), "v"(c));
  return d;
#endif
}

// ---------------------------------------------------------------------------
// Kernel 1: bounds1 = IBP(W1, b1, bounds0).  One block per output row.
// ---------------------------------------------------------------------------
__global__ __launch_bounds__(256) void k_bounds1(
    const float* __restrict__ W1, const float* __restrict__ b1,
    const float* __restrict__ bounds0,
    float* __restrict__ lo1, float* __restrict__ up1) {
  const int i = blockIdx.x;
  const int t = threadIdx.x;
  const float* row = W1 + (size_t)i * N_IN;
  float lo = 0.f, up = 0.f;
  for (int j = t; j < N_IN; j += 256) {
    float w = row[j];
    float l = bounds0[2 * j], u = bounds0[2 * j + 1];
    float p = fmaxf(w, 0.f), n = fminf(w, 0.f);
    lo += p * l + n * u;
    up += p * u + n * l;
  }
  __shared__ float sl[256], su[256];
  sl[t] = lo; su[t] = up;
  __syncthreads();
  for (int s = 128; s > 0; s >>= 1) {
    if (t < s) { sl[t] += sl[t + s]; su[t] += su[t + s]; }
    __syncthreads();
  }
  if (t == 0) { lo1[i] = sl[0] + b1[i]; up1[i] = su[0] + b1[i]; }
}

// ---------------------------------------------------------------------------
// Kernel 2: naive = IBP(W2, b2, bounds1)  fused with  UBV = Up@b1 + b2,
// LBV = Un@b1 + b2 (single pass over W2 rows).
// ---------------------------------------------------------------------------
__global__ __launch_bounds__(256) void k_naive_vec(
    const float* __restrict__ W2, const float* __restrict__ b1v,
    const float* __restrict__ b2v,
    const float* __restrict__ lo1, const float* __restrict__ up1,
    float* __restrict__ naive_lo, float* __restrict__ naive_up,
    float* __restrict__ UBV, float* __restrict__ LBV) {
  const int i = blockIdx.x;
  const int t = threadIdx.x;
  const float* row = W2 + (size_t)i * N_HID;
  float nlo = 0.f, nup = 0.f, ub = 0.f, lb = 0.f;
  for (int j = t; j < N_HID; j += 256) {
    float w = row[j];
    float p = fmaxf(w, 0.f), n = fminf(w, 0.f);
    float l = lo1[j], u = up1[j], b = b1v[j];
    nlo += p * l + n * u;
    nup += p * u + n * l;
    ub  += p * b;
    lb  += n * b;
  }
  __shared__ float s0[256], s1[256], s2[256], s3[256];
  s0[t] = nlo; s1[t] = nup; s2[t] = ub; s3[t] = lb;
  __syncthreads();
  for (int s = 128; s > 0; s >>= 1) {
    if (t < s) { s0[t] += s0[t+s]; s1[t] += s1[t+s]; s2[t] += s2[t+s]; s3[t] += s3[t+s]; }
    __syncthreads();
  }
  if (t == 0) {
    float b2i = b2v[i];
    naive_lo[i] = s0[0] + b2i;
    naive_up[i] = s1[0] + b2i;
    UBV[i]      = s2[0] + b2i;
    LBV[i]      = s3[0] + b2i;
  }
}

// ---------------------------------------------------------------------------
// Kernel 3: fused WMMA GEMM + concretization.
//   S = W2 @ W1,  A = |W2| @ |W1|   (both accumulated with v_wmma_f32_16x16x4)
//   LBM = (S-A)/2, UBM = (S+A)/2 applied element-wise against bounds0 in the
//   epilogue, reduced over the 64-wide j tile into per-(jb, i) partials.
// Grid: (N_OUT/128, N_IN/64); block 256 = 8 waves; wave tile 16(M) x 64(N).
// ---------------------------------------------------------------------------
__global__ __launch_bounds__(256) void k_backsub_wmma(
    const float* __restrict__ W1, const float* __restrict__ W2,
    const float* __restrict__ bounds0,
    float* __restrict__ part_lo, float* __restrict__ part_up) {
  const int i0 = blockIdx.x * 128;
  const int j0 = blockIdx.y * 64;
  const int jb = blockIdx.y;
  const int t = threadIdx.x;
  const int wid   = t >> 5;       // 0..7 : wave index (M strips of 16)
  const int lane  = t & 31;
  const int lhalf = lane >> 4;    // 0/1
  const int lmod  = lane & 15;

  __shared__ float As[128 * 16];  // W2 tile  [m][k], k contiguous
  __shared__ float Bs[16 * 64];   // W1 tile  [k][n], n contiguous

  v8f accS[4], accA[4];
  {
    v8f z = {};
    for (int q = 0; q < 4; ++q) { accS[q] = z; accA[q] = z; }
  }

  // Preload this lane's input-box values for its 4 N-subtiles.
  float l0v[4], u0v[4];
#pragma unroll
  for (int nt = 0; nt < 4; ++nt) {
    int j = j0 + nt * 16 + lmod;
    l0v[nt] = bounds0[2 * j];
    u0v[nt] = bounds0[2 * j + 1];
  }

  for (int kb = 0; kb < N_HID; kb += 16) {
    // Stage W2 tile: 128 rows x 16 k.  8 floats (2x float4) per thread.
    {
      int r  = t >> 1;
      int kk = (t & 1) * 8;
      const float4* src =
          (const float4*)(W2 + (size_t)(i0 + r) * N_HID + kb + kk);
      float4 v0 = src[0];
      float4 v1 = src[1];
      float4* dst = (float4*)(As + r * 16 + kk);
      dst[0] = v0;
      dst[1] = v1;
    }
    // Stage W1 tile: 16 k rows x 64 j.  1 float4 per thread.
    {
      int kr = t >> 4;
      int jc = (t & 15) * 4;
      float4 v = *(const float4*)(W1 + (size_t)(kb + kr) * N_IN + j0 + jc);
      *(float4*)(Bs + kr * 64 + jc) = v;
    }
    __syncthreads();

#pragma unroll
    for (int kc = 0; kc < 4; ++kc) {
      // A fragment (16x4 f32): lane -> M=lmod, VGPR v -> K = kc*4 + 2*lhalf + v
      const int kbase = kc * 4 + lhalf * 2;
      const int m = wid * 16 + lmod;
      v2f a = *(const v2f*)(As + m * 16 + kbase);
      v2f aa;
      aa.x = __builtin_fabsf(a.x);
      aa.y = __builtin_fabsf(a.y);
#pragma unroll
      for (int nt = 0; nt < 4; ++nt) {
        // B fragment (4x16 f32): lane -> N=lmod, VGPR v -> K = kc*4 + 2*lhalf + v
        const int n = nt * 16 + lmod;
        v2f b;
        b.x = Bs[(kbase + 0) * 64 + n];
        b.y = Bs[(kbase + 1) * 64 + n];
        v2f bb;
        bb.x = __builtin_fabsf(b.x);
        bb.y = __builtin_fabsf(b.y);
        accS[nt] = wmma_f32_16x16x4(a,  b,  accS[nt]);
        accA[nt] = wmma_f32_16x16x4(aa, bb, accA[nt]);
      }
    }
    __syncthreads();
  }

  // Epilogue: element (lane, VGPR r) of C/D holds (i = r + 8*lhalf, j = lmod)
  // within the 16x16 subtile.  Fold the box bounds, reduce across the 16
  // j-lanes of each half, and emit one partial per (jb, i).
#pragma unroll
  for (int r = 0; r < 8; ++r) {
    float loS = 0.f, upS = 0.f;
#pragma unroll
    for (int nt = 0; nt < 4; ++nt) {
      float S  = accS[nt][r];
      float Ab = accA[nt][r];
      float LBM = 0.5f * (S - Ab);
      float UBM = 0.5f * (S + Ab);
      loS += fmaxf(LBM, 0.f) * l0v[nt] + fminf(LBM, 0.f) * u0v[nt];
      upS += fmaxf(UBM, 0.f) * u0v[nt] + fminf(UBM, 0.f) * l0v[nt];
    }
#pragma unroll
    for (int off = 1; off < 16; off <<= 1) {   // stays within each 16-lane half
      loS += __shfl_xor(loS, off, 32);
      upS += __shfl_xor(upS, off, 32);
    }
    if (lmod == 0) {
      int i = i0 + wid * 16 + r + 8 * lhalf;
      part_lo[(size_t)jb * N_OUT + i] = loS;
      part_up[(size_t)jb * N_OUT + i] = upS;
    }
  }
}

// ---------------------------------------------------------------------------
// Kernel 4: deterministic reduction over j-blocks + final tighten-vs-naive.
// ---------------------------------------------------------------------------
__global__ __launch_bounds__(256) void k_finalize(
    const float* __restrict__ part_lo, const float* __restrict__ part_up,
    const float* __restrict__ LBV, const float* __restrict__ UBV,
    const float* __restrict__ naive_lo, const float* __restrict__ naive_up,
    float* __restrict__ out) {
  int i = blockIdx.x * blockDim.x + threadIdx.x;
  if (i >= N_OUT) return;
  float lo = 0.f, up = 0.f;
  for (int jb = 0; jb < NJB; ++jb) {
    lo += part_lo[(size_t)jb * N_OUT + i];
    up += part_up[(size_t)jb * N_OUT + i];
  }
  lo += LBV[i];
  up += UBV[i];
  out[2 * i]     = fmaxf(lo, naive_lo[i]);
  out[2 * i + 1] = fminf(up, naive_up[i]);
}

extern "C" void kernel_launch(void* const* d_in, const int* in_sizes, int n_in,
                              void* d_out, int out_size, void* d_ws, size_t ws_size,
                              hipStream_t stream) {
  const float* bounds0 = (const float*)d_in[0];  // [4096,2]
  const float* W1      = (const float*)d_in[1];  // [4096,4096]
  const float* b1      = (const float*)d_in[2];  // [4096]
  const float* W2      = (const float*)d_in[3];  // [4096,4096]
  const float* b2      = (const float*)d_in[4];  // [4096]
  float* out = (float*)d_out;                    // [4096,2]

  float* ws = (float*)d_ws;
  float* lo1      = ws;                 // 4096
  float* up1      = ws + 4096;          // 4096
  float* naive_lo = ws + 8192;          // 4096
  float* naive_up = ws + 12288;         // 4096
  float* UBV      = ws + 16384;         // 4096
  float* LBV      = ws + 20480;         // 4096
  float* part_lo  = ws + 24576;         // 64*4096
  float* part_up  = part_lo + (size_t)NJB * N_OUT;

  k_bounds1<<<N_HID, 256, 0, stream>>>(W1, b1, bounds0, lo1, up1);
  k_naive_vec<<<N_OUT, 256, 0, stream>>>(W2, b1, b2, lo1, up1,
                                         naive_lo, naive_up, UBV, LBV);
  dim3 grid(N_OUT / 128, N_IN / 64);
  k_backsub_wmma<<<grid, 256, 0, stream>>>(W1, W2, bounds0, part_lo, part_up);
  k_finalize<<<(N_OUT + 255) / 256, 256, 0, stream>>>(
      part_lo, part_up, LBV, UBV, naive_lo, naive_up, out);
}